// MultiFrameModule_16870631538824
// MI455X (gfx1250) — compile-verified
//
#include <hip/hip_runtime.h>

// Problem shape from reference: spec [B,C,T,F], Rxx [B,C,T,F,5,5], rss [B,C,T,F,5]
#define FRAME   5
#define DIM_B   4
#define DIM_C   2
#define DIM_T   1000
#define DIM_F   257
#define NSITES  (DIM_B * DIM_C * DIM_T * DIM_F)   // 2,056,000

#define BLOCK        256
#define SITE_FLOATS  (FRAME * FRAME)              // 25 floats = 100 B per site
#define LDS_FLOATS   (BLOCK * SITE_FLOATS)        // 6400 floats = 25,600 B per block

#if defined(__gfx1250__) && __has_builtin(__builtin_amdgcn_global_load_async_to_lds_b128)
#define USE_ASYNC_LDS 1
#else
#define USE_ASYNC_LDS 0
#endif

#if USE_ASYNC_LDS
typedef int v4i __attribute__((vector_size(16)));
typedef __attribute__((address_space(1))) v4i* gas_ptr_t;   // global (prints as __device__)
typedef __attribute__((address_space(3))) v4i* las_ptr_t;   // LDS    (prints as __shared__)
#endif

__global__ __launch_bounds__(BLOCK)
void mf_solve_apply(const float* __restrict__ spec,
                    const float* __restrict__ Rxx,
                    const float* __restrict__ rss,
                    float* __restrict__ out)
{
    __shared__ float ldsM[LDS_FLOATS];            // staged Rxx tile (100 B / site)

    const int tid  = threadIdx.x;
    const int base = blockIdx.x * BLOCK;
    const int s    = base + tid;

    const int rem    = NSITES - base;
    const int nsites = (rem < BLOCK) ? rem : BLOCK;
    const int ndw    = nsites * SITE_FLOATS;      // dwords to stage this block
    const float* gsrc = Rxx + (size_t)base * SITE_FLOATS;

#if USE_ASYNC_LDS
    {
        // Cooperative, fully coalesced async copy: 16 B per lane, unit stride.
        const int nb128 = ndw >> 2;               // number of 16-byte chunks
        const char* gs = (const char*)gsrc;
        char* ls = (char*)ldsM;
        for (int i = tid; i < nb128; i += BLOCK) {
            __builtin_amdgcn_global_load_async_to_lds_b128(
                (gas_ptr_t)(gs + 16 * i),
                (las_ptr_t)(ls + 16 * i),
                /*offset=*/0, /*cpol=*/0);
        }
        // dword tail (always 0 here: nsites is 256 or 64, both %4==0; kept for safety)
        for (int i = (nb128 << 2) + tid; i < ndw; i += BLOCK) {
            ldsM[i] = gsrc[i];
        }
        asm volatile("s_wait_asynccnt 0" ::: "memory");
    }
#else
    // Fallback staging with plain loads (still coalesced).
    for (int i = tid; i < ndw; i += BLOCK) ldsM[i] = gsrc[i];
#endif
    __syncthreads();

    if (s >= NSITES) return;

    // Read this site's 5x5 from LDS. Dword index = 25*tid + k; stride 25 is odd
    // (gcd(25,64)==1) -> conflict-free across the 64 banks.
    float M[FRAME][FRAME];
    {
        const float* Mp = &ldsM[tid * SITE_FLOATS];
        #pragma unroll
        for (int i = 0; i < FRAME; ++i)
            #pragma unroll
            for (int j = 0; j < FRAME; ++j)
                M[i][j] = Mp[i * FRAME + j];
    }

    float r[FRAME];
    {
        const float* rp = rss + (size_t)s * FRAME;
        #pragma unroll
        for (int n = 0; n < FRAME; ++n) r[n] = rp[n];
    }

    // Tikhonov: M += (trace*1e-8 + 1e-7) * I
    float tr = 0.f;
    #pragma unroll
    for (int i = 0; i < FRAME; ++i) tr += M[i][i];
    const float eps = fmaf(tr, 1e-8f, 1e-7f);
    #pragma unroll
    for (int i = 0; i < FRAME; ++i) M[i][i] += eps;

    // Gaussian elimination without pivoting (SPD after regularization).
    float dinv[FRAME];
    #pragma unroll
    for (int k = 0; k < FRAME; ++k) {
        const float inv = __builtin_amdgcn_rcpf(M[k][k]);
        dinv[k] = inv;
        #pragma unroll
        for (int i = k + 1; i < FRAME; ++i) {
            const float fct = M[i][k] * inv;
            #pragma unroll
            for (int j = k + 1; j < FRAME; ++j)
                M[i][j] = fmaf(-fct, M[k][j], M[i][j]);
            r[i] = fmaf(-fct, r[k], r[i]);
        }
    }
    // Back substitution.
    float x[FRAME];
    #pragma unroll
    for (int k = FRAME - 1; k >= 0; --k) {
        float acc = r[k];
        #pragma unroll
        for (int j = k + 1; j < FRAME; ++j)
            acc = fmaf(-M[k][j], x[j], acc);
        x[k] = acc * dinv[k];
    }

    // Apply coefficients to time-shifted spec (pad_before=2, pad_after=2):
    // spec_unf[...,n] = spec[b,c,t+n-2,f]  -> flat index s + (n-2)*F, guarded on t.
    const int t = (s / DIM_F) % DIM_T;
    float acc = 0.f;
    #pragma unroll
    for (int n = 0; n < FRAME; ++n) {
        const int tt = t + n - 2;
        float sv = 0.f;
        if (tt >= 0 && tt < DIM_T) sv = spec[s + (n - 2) * DIM_F];
        acc = fmaf(sv, x[n], acc);
    }
    out[s] = acc;
}

extern "C" void kernel_launch(void* const* d_in, const int* in_sizes, int n_in,
                              void* d_out, int out_size, void* d_ws, size_t ws_size,
                              hipStream_t stream) {
    (void)in_sizes; (void)n_in; (void)out_size; (void)d_ws; (void)ws_size;
    const float* spec = (const float*)d_in[0];
    const float* Rxx  = (const float*)d_in[1];
    const float* rss  = (const float*)d_in[2];
    float* out        = (float*)d_out;

    const int grid = (NSITES + BLOCK - 1) / BLOCK;   // 8032 blocks of 256 (8 waves)
    mf_solve_apply<<<grid, BLOCK, 0, stream>>>(spec, Rxx, rss, out);
}